// ANEDecoderLayer_38104949850604
// MI455X (gfx1250) — compile-verified
//
#include <hip/hip_runtime.h>
#include <math.h>

#define HIDN 4096
#define NH   32
#define NKV  8
#define DH   128
#define IMLP 12288
#define SSEQ 8192
#define EPSF 1e-6f
#define NCHUNK 32
#define CHUNK  (SSEQ / NCHUNK)   // 256
#define QKSCALE 0.08838834764831843f  // 128^-0.5

typedef float v2f __attribute__((ext_vector_type(2)));
typedef float v4f __attribute__((ext_vector_type(4)));
typedef float v8f __attribute__((ext_vector_type(8)));

__device__ __forceinline__ float wave_sum(float v) {
#pragma unroll
  for (int off = 16; off >= 1; off >>= 1) v += __shfl_xor(v, off, 32);
  return v;
}

// ---------------------------------------------------------------- cache copy
__global__ void __launch_bounds__(256)
copy_cache_kernel(const v4f* __restrict__ kc, const v4f* __restrict__ vc,
                  v4f* __restrict__ ok, v4f* __restrict__ ov) {
  size_t i = (size_t)blockIdx.x * blockDim.x + threadIdx.x;
  const size_t n = (size_t)NKV * SSEQ * DH / 4;   // 2,097,152 v4f per cache
  if (i < n) ok[i] = __builtin_nontemporal_load(&kc[i]);
  else if (i < 2 * n) ov[i - n] = __builtin_nontemporal_load(&vc[i - n]);
}

// ---------------------------------------------------------------- RMSNorm (vector)
__global__ void __launch_bounds__(256)
rmsnorm_kernel(const float* __restrict__ x, const float* __restrict__ w,
               float* __restrict__ out, int n) {
  __shared__ float red[8];
  __shared__ float smean;
  float s = 0.f;
  for (int i = threadIdx.x; i < n; i += blockDim.x) { float v = x[i]; s += v * v; }
  s = wave_sum(s);
  int wid = threadIdx.x >> 5, lane = threadIdx.x & 31;
  if (lane == 0) red[wid] = s;
  __syncthreads();
  if (threadIdx.x == 0) {
    float t = 0.f;
    for (int i = 0; i < (int)(blockDim.x >> 5); i++) t += red[i];
    smean = t / (float)n;
  }
  __syncthreads();
  float inv = rsqrtf(smean + EPSF);
  for (int i = threadIdx.x; i < n; i += blockDim.x) out[i] = x[i] * inv * w[i];
}

// ---------------------------------------------------------------- GEMV core
__device__ __forceinline__ float gemv_dot(const float* __restrict__ Wrow,
                                          const float* __restrict__ hsh,
                                          int ncols4, int lane) {
  float acc = 0.f;
  const v4f* w4 = (const v4f*)Wrow;
  const v4f* h4 = (const v4f*)hsh;
  for (int c = lane; c < ncols4; c += 32) {
    v4f wv = __builtin_nontemporal_load(&w4[c]);   // single-use weights: NT
    v4f hv = h4[c];
    acc += wv.x * hv.x + wv.y * hv.y + wv.z * hv.z + wv.w * hv.w;
  }
  return wave_sum(acc);
}

__global__ void __launch_bounds__(256)
gemv_qkv_kernel(const float* __restrict__ Wq, const float* __restrict__ Wk,
                const float* __restrict__ Wv, const float* __restrict__ hin,
                float* __restrict__ out) {
  __shared__ __align__(16) float hs[HIDN];
  for (int i = threadIdx.x; i < HIDN; i += blockDim.x) hs[i] = hin[i];
  __syncthreads();
  int lane = threadIdx.x & 31;
  int row = blockIdx.x * (blockDim.x >> 5) + (threadIdx.x >> 5);  // 0..6143
  const float* Wrow;
  if (row < 4096)      Wrow = Wq + (size_t)row * HIDN;
  else if (row < 5120) Wrow = Wk + (size_t)(row - 4096) * HIDN;
  else                 Wrow = Wv + (size_t)(row - 5120) * HIDN;
  float r = gemv_dot(Wrow, hs, HIDN / 4, lane);
  if (lane == 0) out[row] = r;
}

__global__ void __launch_bounds__(256)
gemv_wo_kernel(const float* __restrict__ Wo, const float* __restrict__ attn,
               const float* __restrict__ resid, float* __restrict__ out) {
  __shared__ __align__(16) float hs[NH * DH];
  for (int i = threadIdx.x; i < NH * DH; i += blockDim.x) hs[i] = attn[i];
  __syncthreads();
  int lane = threadIdx.x & 31;
  int row = blockIdx.x * (blockDim.x >> 5) + (threadIdx.x >> 5);
  float r = gemv_dot(Wo + (size_t)row * (NH * DH), hs, (NH * DH) / 4, lane);
  if (lane == 0) out[row] = resid[row] + r;
}

__global__ void __launch_bounds__(256)
gemv_gu_kernel(const float* __restrict__ Wg, const float* __restrict__ Wu,
               const float* __restrict__ h2, float* __restrict__ gu) {
  __shared__ __align__(16) float hs[HIDN];
  for (int i = threadIdx.x; i < HIDN; i += blockDim.x) hs[i] = h2[i];
  __syncthreads();
  int lane = threadIdx.x & 31;
  int row = blockIdx.x * (blockDim.x >> 5) + (threadIdx.x >> 5);  // 0..12287
  const v4f* wg4 = (const v4f*)(Wg + (size_t)row * HIDN);
  const v4f* wu4 = (const v4f*)(Wu + (size_t)row * HIDN);
  const v4f* h4 = (const v4f*)hs;
  float ag = 0.f, au = 0.f;
  for (int c = lane; c < HIDN / 4; c += 32) {
    v4f g4 = __builtin_nontemporal_load(&wg4[c]);
    v4f u4 = __builtin_nontemporal_load(&wu4[c]);
    v4f hv = h4[c];
    ag += g4.x * hv.x + g4.y * hv.y + g4.z * hv.z + g4.w * hv.w;
    au += u4.x * hv.x + u4.y * hv.y + u4.z * hv.z + u4.w * hv.w;
  }
  ag = wave_sum(ag);
  au = wave_sum(au);
  if (lane == 0) {
    float s = ag / (1.f + __expf(-ag));   // silu
    gu[row] = s * au;
  }
}

__global__ void __launch_bounds__(256)
gemv_wd_kernel(const float* __restrict__ Wd, const float* __restrict__ gu,
               const float* __restrict__ h1a, float* __restrict__ out) {
  __shared__ __align__(16) float hs[IMLP];   // 48 KB
  for (int i = threadIdx.x; i < IMLP; i += blockDim.x) hs[i] = gu[i];
  __syncthreads();
  int lane = threadIdx.x & 31;
  int row = blockIdx.x * (blockDim.x >> 5) + (threadIdx.x >> 5);
  float r = gemv_dot(Wd + (size_t)row * IMLP, hs, IMLP / 4, lane);
  if (lane == 0) out[row] = h1a[row] + r;
}

// ------------------------------------------------- QK RMSNorm + RoPE + cache row write
__global__ void __launch_bounds__(128)
qknorm_rope_kernel(float* __restrict__ qkv, const float* __restrict__ cosv,
                   const float* __restrict__ sinv, const float* __restrict__ qw,
                   const float* __restrict__ kw, const float* __restrict__ posp,
                   float* __restrict__ outk, float* __restrict__ outv) {
  int r = blockIdx.x, t = threadIdx.x;
  int ipos = (int)posp[0];
  if (r >= 40) {                 // v rows: raw copy into cache at pos
    int g = r - 40;
    outv[((size_t)(g * SSEQ + ipos)) * DH + t] = qkv[5120 + g * DH + t];
    return;
  }
  float* row;
  const float* w;
  if (r < 32) { row = qkv + r * DH;              w = qw; }
  else        { row = qkv + 4096 + (r - 32) * DH; w = kw; }
  __shared__ float red[4];
  __shared__ float nrm[DH];
  __shared__ float smean;
  float x = row[t];
  float s = wave_sum(x * x);
  int wid = t >> 5, lane = t & 31;
  if (lane == 0) red[wid] = s;
  __syncthreads();
  if (t == 0) smean = (red[0] + red[1] + red[2] + red[3]) / (float)DH;
  __syncthreads();
  float nv = x * rsqrtf(smean + EPSF) * w[t];
  nrm[t] = nv;
  __syncthreads();
  float rh = (t < 64) ? -nrm[t + 64] : nrm[t - 64];
  float o = nv * cosv[t] + rh * sinv[t];
  row[t] = o;
  if (r >= 32) {
    int g = r - 32;
    outk[((size_t)(g * SSEQ + ipos)) * DH + t] = o;
  }
}

// ---------------------------------------------------------------- attention (WMMA f32)
// One wave per (kv-head g, position chunk c). Flash-decoding partials.
// Fragment layouts per CDNA5 ISA §7.12.2 (f32 16x16x4):
//   A/B: lane L, vgpr r  ->  element [L&15][(L>>4)*2 + r]
//   C/D: lane L, comp j  ->  element [j + 8*(L>>4)][L&15]
__global__ void __launch_bounds__(32)
attn_partial_kernel(const float* __restrict__ qv, const float* __restrict__ kc,
                    const float* __restrict__ vc, const float* __restrict__ posp,
                    float* __restrict__ pm, float* __restrict__ pl,
                    float* __restrict__ po) {
  int g = blockIdx.x / NCHUNK;
  int c = blockIdx.x % NCHUNK;
  int s0 = c * CHUNK;
  int ipos = (int)posp[0];
  int lane = threadIdx.x;
  int lo = lane & 15, hi = lane >> 4;
  int pbase = (g * NCHUNK + c) * 4;

  if (s0 > ipos) {   // fully masked chunk: exp underflows to exactly 0 in fp32
    if (lane == 0)
      for (int j = 0; j < 4; j++) { pm[pbase + j] = -1e30f; pl[pbase + j] = 0.f; }
    for (int i = lane; i < 4 * DH; i += 32) po[(size_t)pbase * DH + i] = 0.f;
    return;
  }

  __shared__ __align__(16) float qs[4 * DH];
  __shared__ __align__(16) float Ks[16 * DH];
  __shared__ __align__(16) float Vs[16 * DH];
  __shared__ __align__(16) float Ps[16 * 16];

  for (int i = lane; i < 4 * DH; i += 32) qs[i] = qv[g * 4 * DH + i];
  __syncthreads();

  float mrun[4], lrun[4];
  v8f acc[8];
#pragma unroll
  for (int j = 0; j < 4; j++) { mrun[j] = -1e30f; lrun[j] = 0.f; }
#pragma unroll
  for (int db = 0; db < 8; db++) { v8f z = {}; acc[db] = z; }

  const float* kbase = kc + (size_t)(g * SSEQ + s0) * DH;
  const float* vbase = vc + (size_t)(g * SSEQ + s0) * DH;

  for (int t = 0; t < CHUNK / 16; t++) {
    int stile = s0 + t * 16;
    if (stile > ipos) break;                     // uniform branch

    const v4f* k4 = (const v4f*)(kbase + (size_t)t * 16 * DH);
    const v4f* v4 = (const v4f*)(vbase + (size_t)t * 16 * DH);
    v4f* Ks4 = (v4f*)Ks;
    v4f* Vs4 = (v4f*)Vs;
    for (int i = lane; i < (16 * DH) / 4; i += 32) { Ks4[i] = k4[i]; Vs4[i] = v4[i]; }
    __syncthreads();

    // ---- scores: D(16x16) += Q(16x4) x K^T(4x16), accumulate over d
    v8f sc = {};
#pragma unroll
    for (int d = 0; d < 32; d++) {
      int col = 4 * d + 2 * hi;
      v2f a = {};
      if (lo < 4) a = *(const v2f*)&qs[lo * DH + col];     // rows 4..15 are zero
      v2f b = *(const v2f*)&Ks[lo * DH + col];
      sc = __builtin_amdgcn_wmma_f32_16x16x4_f32(false, a, false, b,
                                                 (short)0, sc, false, false);
    }
    float pen = ((stile + lo) > ipos) ? -10000.0f : 0.0f;
#pragma unroll
    for (int j = 0; j < 8; j++) sc[j] = sc[j] * QKSCALE + pen;

    // ---- online softmax over the 4 real rows
    float p[4], f[4];
#pragma unroll
    for (int j = 0; j < 4; j++) {
      float mv = sc[j];
      for (int off = 8; off >= 1; off >>= 1) mv = fmaxf(mv, __shfl_xor(mv, off, 32));
      mv = __shfl(mv, 0, 32);
      float mnew = fmaxf(mrun[j], mv);
      f[j] = __expf(mrun[j] - mnew);
      p[j] = __expf(sc[j] - mnew);
      float sum = p[j];
      for (int off = 8; off >= 1; off >>= 1) sum += __shfl_xor(sum, off, 32);
      sum = __shfl(sum, 0, 32);
      lrun[j] = lrun[j] * f[j] + sum;
      mrun[j] = mnew;
    }
    __syncthreads();      // WAR on Ps vs previous tile's reads
#pragma unroll
    for (int j = 0; j < 8; j++) {
      float val = 0.f;
      if (j < 4 && hi == 0) val = p[j];
      Ps[(j + 8 * hi) * 16 + lo] = val;   // rows 4..15 zeroed
    }
    __syncthreads();

#pragma unroll
    for (int db = 0; db < 8; db++)
#pragma unroll
      for (int j = 0; j < 4; j++) acc[db][j] *= f[j];

    // ---- o += P(16x16-pos) x V(16-pos x 128), K-dim = positions in steps of 4
#pragma unroll
    for (int ks = 0; ks < 4; ks++) {
      v2f a = *(const v2f*)&Ps[lo * 16 + 4 * ks + 2 * hi];
      int r0 = 4 * ks + 2 * hi;
#pragma unroll
      for (int db = 0; db < 8; db++) {
        v2f b;
        b.x = Vs[r0 * DH + db * 16 + lo];
        b.y = Vs[(r0 + 1) * DH + db * 16 + lo];
        acc[db] = __builtin_amdgcn_wmma_f32_16x16x4_f32(false, a, false, b,
                                                        (short)0, acc[db], false, false);
      }
    }
    __syncthreads();      // before next tile overwrites Ks/Vs
  }

  if (lane == 0)
    for (int j = 0; j < 4; j++) { pm[pbase + j] = mrun[j]; pl[pbase + j] = lrun[j]; }
  if (hi == 0) {
#pragma unroll
    for (int db = 0; db < 8; db++)
#pragma unroll
      for (int j = 0; j < 4; j++)
        po[(size_t)(pbase + j) * DH + db * 16 + lo] = acc[db][j];
  }
}

__global__ void __launch_bounds__(128)
attn_combine_kernel(const float* __restrict__ pm, const float* __restrict__ pl,
                    const float* __restrict__ po, float* __restrict__ attn) {
  int h = blockIdx.x;               // 0..31
  int g = h >> 2, j = h & 3;
  int t = threadIdx.x;              // 0..127
  float M = -1e30f;
  for (int c = 0; c < NCHUNK; c++) M = fmaxf(M, pm[(g * NCHUNK + c) * 4 + j]);
  float lt = 0.f, num = 0.f;
  for (int c = 0; c < NCHUNK; c++) {
    int pb = (g * NCHUNK + c) * 4 + j;
    float w = __expf(pm[pb] - M);
    lt += w * pl[pb];
    num += w * po[(size_t)pb * DH + t];
  }
  attn[h * DH + t] = num / lt;
}

// ---------------------------------------------------------------- launch
extern "C" void kernel_launch(void* const* d_in, const int* in_sizes, int n_in,
                              void* d_out, int out_size, void* d_ws, size_t ws_size,
                              hipStream_t stream) {
  const float* x      = (const float*)d_in[0];   // hidden (4096)
  const float* posp   = (const float*)d_in[1];
  const float* cosv   = (const float*)d_in[2];
  const float* sinv   = (const float*)d_in[3];
  const float* kcache = (const float*)d_in[4];
  const float* vcache = (const float*)d_in[5];
  const float* Wq     = (const float*)d_in[6];
  const float* Wk     = (const float*)d_in[7];
  const float* Wv     = (const float*)d_in[8];
  const float* Wo     = (const float*)d_in[9];
  const float* Wg     = (const float*)d_in[10];
  const float* Wu     = (const float*)d_in[11];
  const float* Wd     = (const float*)d_in[12];
  const float* qnw    = (const float*)d_in[13];
  const float* knw    = (const float*)d_in[14];
  const float* ln1    = (const float*)d_in[15];
  const float* ln2    = (const float*)d_in[16];

  float* out   = (float*)d_out;
  float* out_k = out + HIDN;                              // 8*8192*128
  float* out_v = out_k + (size_t)NKV * SSEQ * DH;

  float* ws = (float*)d_ws;
  float* h1n   = ws;                 // 4096
  float* qkv   = ws + 4096;          // 6144 (q:0..4095, k:4096..5119, v:5120..6143)
  float* attn  = ws + 10240;         // 4096
  float* h1a   = ws + 14336;         // 4096
  float* h2n   = ws + 18432;         // 4096
  float* gu    = ws + 22528;         // 12288
  float* pmb   = ws + 34816;         // 8*32*4 = 1024
  float* plb   = ws + 35840;         // 1024
  float* pob   = ws + 36864;         // 8*32*4*128 = 131072

  // 1) functional KV cache copy-out (row at pos overwritten later)
  copy_cache_kernel<<<16384, 256, 0, stream>>>(
      (const v4f*)kcache, (const v4f*)vcache, (v4f*)out_k, (v4f*)out_v);
  // 2) pre-attention RMSNorm
  rmsnorm_kernel<<<1, 256, 0, stream>>>(x, ln1, h1n, HIDN);
  // 3) fused QKV GEMV (6144 rows)
  gemv_qkv_kernel<<<768, 256, 0, stream>>>(Wq, Wk, Wv, h1n, qkv);
  // 4) QK RMSNorm + RoPE + write new k,v row into the output caches
  qknorm_rope_kernel<<<48, 128, 0, stream>>>(qkv, cosv, sinv, qnw, knw, posp,
                                             out_k, out_v);
  // 5) flash-decode attention partials (WMMA f32 16x16x4)
  attn_partial_kernel<<<NKV * NCHUNK, 32, 0, stream>>>(qkv, out_k, out_v, posp,
                                                       pmb, plb, pob);
  // 6) combine partials
  attn_combine_kernel<<<NH, 128, 0, stream>>>(pmb, plb, pob, attn);
  // 7) output projection + residual
  gemv_wo_kernel<<<512, 256, 0, stream>>>(Wo, attn, x, h1a);
  // 8) pre-MLP RMSNorm
  rmsnorm_kernel<<<1, 256, 0, stream>>>(h1a, ln2, h2n, HIDN);
  // 9) gate/up GEMV + SwiGLU (12288 rows)
  gemv_gu_kernel<<<1536, 256, 0, stream>>>(Wg, Wu, h2n, gu);
  // 10) down projection + residual -> hid output
  gemv_wd_kernel<<<512, 256, 0, stream>>>(Wd, gu, h1a, out);
}